// SortPooling_49289044689301
// MI455X (gfx1250) — compile-verified
//
#include <hip/hip_runtime.h>
#include <stdint.h>

typedef int v4i __attribute__((vector_size(16)));

#define LDS_AS __attribute__((address_space(3)))

#if __has_builtin(__builtin_amdgcn_global_load_async_to_lds_b128)
#define HAVE_ASYNC_BUILTIN 1
#else
#define HAVE_ASYNC_BUILTIN 0
#endif

// Async copy of 16 bytes global -> LDS (gfx1250 GLOBAL_LOAD_ASYNC_TO_LDS_B128).
__device__ __forceinline__ void async_copy16(const float* g, float* l) {
#if HAVE_ASYNC_BUILTIN
    __builtin_amdgcn_global_load_async_to_lds_b128(
        (v4i*)g, (v4i*)l, /*offset=*/0, /*cpol=*/0);
#else
    uint32_t lds_off = (uint32_t)(uintptr_t)(LDS_AS void*)l;
    uint64_t gaddr   = (uint64_t)(uintptr_t)g;
    asm volatile("global_load_async_to_lds_b128 %0, %1, off"
                 :: "v"(lds_off), "v"(gaddr) : "memory");
#endif
}

__device__ __forceinline__ void wait_async0() {
#if __has_builtin(__builtin_amdgcn_s_wait_asynccnt)
    __builtin_amdgcn_s_wait_asynccnt(0);
#else
    asm volatile("s_wait_asynccnt 0" ::: "memory");
#endif
}

// ---------------------------------------------------------------------------
// K1: per-row max (== last element of the sorted row). One wave32 per row.
// 256 threads/block = 8 waves = 8 rows. Each lane loads 2 x float4 (coalesced).
// ---------------------------------------------------------------------------
__global__ void rowmax_kernel(const float* __restrict__ feat,
                              float* __restrict__ scores, int nrows) {
    const int tid  = threadIdx.x;
    const int wave = tid >> 5;
    const int lane = tid & 31;
    const int row  = blockIdx.x * 8 + wave;
    if (row >= nrows) return;

    const float4* p = (const float4*)(feat + (size_t)row * 256);
    float4 a = p[lane];
    float4 b = p[lane + 32];
    float m = fmaxf(fmaxf(fmaxf(a.x, a.y), fmaxf(a.z, a.w)),
                    fmaxf(fmaxf(b.x, b.y), fmaxf(b.z, b.w)));
    // wave32 butterfly reduction
    #pragma unroll
    for (int off = 16; off > 0; off >>= 1)
        m = fmaxf(m, __shfl_xor(m, off, 32));
    if (lane == 0) scores[row] = m;
}

// ---------------------------------------------------------------------------
// K2: stable descending top-64 per graph via rank-by-comparison.
// rank(i) = #{ j : s[j] > s[i]  or (s[j] == s[i] and j < i) }   (== lax.top_k
// tie rule: lower index first). Thread with rank < 64 writes its global row
// index at output slot `rank` -> indices already in top_k output order.
// One block per graph, n threads (n = rows per graph, 512 here).
// ---------------------------------------------------------------------------
__global__ void topk_kernel(const float* __restrict__ scores,
                            int* __restrict__ sel, int n, int kkeep) {
    __shared__ float s[1024];
    const int g = blockIdx.x;
    const int t = threadIdx.x;

    s[t] = scores[(size_t)g * n + t];
    __syncthreads();

    const float my = s[t];
    int rank = 0;
    for (int j = 0; j < n; ++j) {
        const float v = s[j];               // LDS broadcast read (uniform addr)
        rank += (v > my) || (v == my && j < t);
    }
    if (rank < kkeep) sel[g * kkeep + rank] = g * n + t;
}

// ---------------------------------------------------------------------------
// K3: gather the selected rows and bitonic-sort each (256 f32, ascending).
// One block (256 threads) per output row. Row staged into LDS with async
// global->LDS b128 copies (lanes 0..63 move 16B each), drained with
// s_wait_asynccnt 0, then a 36-stage bitonic network in LDS.
// ---------------------------------------------------------------------------
__global__ void sort_gather_kernel(const float* __restrict__ feat,
                                   const int* __restrict__ sel,
                                   float* __restrict__ out) {
    __shared__ float s[256];
    const int blk = blockIdx.x;
    const int tid = threadIdx.x;
    const int row = sel[blk];

    if (tid < 64)
        async_copy16(feat + (size_t)row * 256 + tid * 4, &s[tid * 4]);
    wait_async0();
    __syncthreads();

    // bitonic sort, 256 elements, thread-per-element, ascending
    #pragma unroll 1
    for (int k = 2; k <= 256; k <<= 1) {
        #pragma unroll 1
        for (int j = k >> 1; j > 0; j >>= 1) {
            const int ixj = tid ^ j;
            if (ixj > tid) {
                const float a = s[tid];
                const float b = s[ixj];
                const bool up = ((tid & k) == 0);
                if (up ? (a > b) : (a < b)) { s[tid] = b; s[ixj] = a; }
            }
            __syncthreads();
        }
    }

    out[(size_t)blk * 256 + tid] = s[tid];
}

// ---------------------------------------------------------------------------
// launch
// ---------------------------------------------------------------------------
extern "C" void kernel_launch(void* const* d_in, const int* in_sizes, int n_in,
                              void* d_out, int out_size, void* d_ws, size_t ws_size,
                              hipStream_t stream) {
    const float* feat = (const float*)d_in[0];
    float* out = (float*)d_out;

    const int D = 256;
    const int K = 64;
    const int nrows = in_sizes[0] / D;        // 131072
    const int G = out_size / (K * D);         // 256 graphs
    const int n = nrows / G;                  // 512 rows per graph

    float* scores = (float*)d_ws;                         // nrows f32
    int*   sel    = (int*)((char*)d_ws + (size_t)nrows * sizeof(float)); // G*K i32

    rowmax_kernel<<<(nrows + 7) / 8, 256, 0, stream>>>(feat, scores, nrows);
    topk_kernel<<<G, n, 0, stream>>>(scores, sel, n, K);
    sort_gather_kernel<<<G * K, 256, 0, stream>>>(feat, sel, out);
}